// EnergyConditionedEquivariantAtomAttentionLowRank_80805514707449
// MI455X (gfx1250) — compile-verified
//
#include <hip/hip_runtime.h>
#include <math.h>

// ---------------------------------------------------------------------------
// EnergyConditionedEquivariantAtomAttentionLowRank — MI455X (gfx1250) port.
// All matmul-shaped work runs on V_WMMA_F32_16X16X4_F32 (full-precision f32
// matrix cores): problem is L2-resident (~10 MB), so fp32 WMMA is the right
// precision/data path; low-rank algebra removes the 67M-element tp_w tensor.
// Round 2: branch-free clamped addressing (no exec-masked scalar loads),
// v2f (b64) A-fragment loads, pointer-increment indexing, partial unroll.
// ---------------------------------------------------------------------------

typedef float v2f __attribute__((ext_vector_type(2)));
typedef float v8f __attribute__((ext_vector_type(8)));

constexpr int B_    = 8;
constexpr int N_    = 64;
constexpr int NE_   = 128;
constexpr int DE_   = 16;
constexpr int MUL_  = 16;
constexpr int DN_   = 4 * MUL_;   // 64
constexpr int RBF_  = 32;
constexpr int ZEMB_ = 32;
constexpr int HID_  = 128;
constexpr int LAT_  = 64;
constexpr int RANK_ = 8;
constexpr int GIN_  = ZEMB_ + RBF_;      // 64
constexpr int WNUM_ = 4 * MUL_ * MUL_;   // 1024
constexpr int PQC_  = LAT_ + RANK_ * LAT_;           // 576 (P | geom_c*Q)
constexpr float CUTOFF_ = 6.0f;
constexpr float PI_     = 3.14159265358979323846f;

__device__ __forceinline__ float silu_(float x)    { return x / (1.0f + __expf(-x)); }
__device__ __forceinline__ float sigmoid_(float x) { return 1.0f / (1.0f + __expf(-x)); }

// ---------------------------------------------------------------------------
// Generic WMMA f32 GEMM:  C[b] = act(A[b] (RxK) @ W[b] (KxCn) + bias)
// One wave (32 threads) per 16x16 output tile.  A-fragment: lane m=lane&15,
// koff=(lane>>4)*2 holds A[m][k0+koff], A[m][k0+koff+1]; B mirrored; D rows
// j and j+8 across lane halves (ISA 7.12.2 layouts).  K must be %4==0.
// Loads are unconditional (indices clamped); only stores are masked.
// ---------------------------------------------------------------------------
__global__ void gemm_bias_act(const float* __restrict__ A,
                              const float* __restrict__ W,
                              const float* __restrict__ bias,
                              float* __restrict__ C,
                              int R, int K, int Cn,
                              long long sA, long long sB, long long sC,
                              int act) {
  const int lane = threadIdx.x;           // block = 32 = one wave
  const int m    = lane & 15;
  const int koff = (lane >> 4) << 1;
  const float* Ab = A + (long long)blockIdx.z * sA;
  const float* Wb = W + (long long)blockIdx.z * sB;
  float*       Cb = C + (long long)blockIdx.z * sC;
  const int row  = blockIdx.x * 16 + m;
  const int col  = blockIdx.y * 16 + m;
  const int rowc = (row < R)  ? row : (R - 1);   // clamp: loads always legal
  const int colc = (col < Cn) ? col : (Cn - 1);

  const float* ap = Ab + rowc * K + koff;        // contiguous pair -> b64
  const float* bp = Wb + koff * Cn + colc;       // stride-Cn pair
  const int bstep = 4 * Cn;

  v8f acc = {0.f, 0.f, 0.f, 0.f, 0.f, 0.f, 0.f, 0.f};
  #pragma unroll 4
  for (int k0 = 0; k0 < K; k0 += 4) {
    const v2f a = *(const v2f*)ap;
    v2f b;
    b.x = bp[0];
    b.y = bp[Cn];
    acc = __builtin_amdgcn_wmma_f32_16x16x4_f32(false, a, false, b,
                                                (short)0, acc, false, false);
    ap += 4;
    bp += bstep;
  }
  if (col >= Cn) return;
  const float bv = bias ? bias[col] : 0.f;
  const int rbase = blockIdx.x * 16 + ((lane >> 4) << 3);
  #pragma unroll
  for (int j = 0; j < 8; ++j) {
    const int orow = rbase + j;
    if (orow < R) {
      float v = acc[j] + bv;
      if (act == 1)      v = silu_(v);
      else if (act == 2) v = sigmoid_(v);
      Cb[orow * Cn + col] = v;
    }
  }
}

// ---------------------------------------------------------------------------
// K1: per-(b,n) geometry: RBF, z-embedding gather, sh1, invariants, cos-window
// ---------------------------------------------------------------------------
__global__ void geom_kernel(const float* __restrict__ h_full,
                            const int* __restrict__ z,
                            const float* __restrict__ pos,
                            const unsigned char* __restrict__ mask,
                            const int* __restrict__ absp,
                            const float* __restrict__ z_emb,
                            float* __restrict__ geom_in,   // (B*N, 64) = [zr|rr]
                            float* __restrict__ inv,       // (B*N, 32)
                            float* __restrict__ sh1,       // (B*N, 3)
                            float* __restrict__ cwv) {     // (B*N)
  const int bn = blockIdx.x;
  const int b  = bn >> 6;
  const int n  = bn & 63;
  const int t  = threadIdx.x;          // 32 threads
  const int ai = absp[0];

  const float px = pos[bn * 3 + 0] - pos[(b * N_ + ai) * 3 + 0];
  const float py = pos[bn * 3 + 1] - pos[(b * N_ + ai) * 3 + 1];
  const float pz = pos[bn * 3 + 2] - pos[(b * N_ + ai) * 3 + 2];
  const float r  = sqrtf(px * px + py * py + pz * pz);
  const float ri = 1.0f / fmaxf(r, 1e-8f);
  const float ux = px * ri, uy = py * ri, uz = pz * ri;

  // rr: gaussian RBF on clamped r
  const float delta = CUTOFF_ / (RBF_ - 1);
  const float gamma = 1.0f / (delta * delta + 1e-12f);
  const float rc    = fminf(r, CUTOFF_);
  const float ctr   = CUTOFF_ * (float)t / (float)(RBF_ - 1);
  const float dd    = rc - ctr;
  geom_in[bn * GIN_ + ZEMB_ + t] = __expf(-gamma * dd * dd);

  // zr gather
  const int zi = z[bn];
  geom_in[bn * GIN_ + t] = z_emb[zi * ZEMB_ + t];

  if (t < 3) {
    const float s3 = 1.7320508075688772f;
    sh1[bn * 3 + t] = s3 * (t == 0 ? uy : (t == 1 ? uz : ux));  // u[[1,2,0]]
  }
  // invariants of h_full[b,n]
  if (t < MUL_) {
    inv[bn * 32 + t] = h_full[bn * DN_ + t];
  } else {
    const int mm = t - MUL_;
    const float v0 = h_full[bn * DN_ + MUL_ + mm * 3 + 0];
    const float v1 = h_full[bn * DN_ + MUL_ + mm * 3 + 1];
    const float v2 = h_full[bn * DN_ + MUL_ + mm * 3 + 2];
    inv[bn * 32 + t] = sqrtf((v0 * v0 + v1 * v1 + v2 * v2) * (1.0f / 3.0f) + 1e-8f);
  }
  if (t == 0) {
    const bool valid = (mask[bn] != 0) && (n != ai) && (r <= CUTOFF_);
    cwv[bn] = valid ? 0.5f * (cosf(PI_ * r / CUTOFF_) + 1.0f) : 0.0f;
  }
}

// ---------------------------------------------------------------------------
// K1b: assemble per-(b,n) part of score input: [inv_abs | inv_nei | zr | rr]
// ---------------------------------------------------------------------------
__global__ void x1_kernel(const float* __restrict__ inv,
                          const float* __restrict__ geom_in,
                          const int* __restrict__ absp,
                          float* __restrict__ X1) {        // (B*N, 128)
  const int bn = blockIdx.x;
  const int b  = bn >> 6;
  const int t  = threadIdx.x;   // 128 threads
  const int ai = absp[0];
  float v;
  if (t < 32)       v = inv[(b * N_ + ai) * 32 + t];
  else if (t < 64)  v = inv[bn * 32 + (t - 32)];
  else              v = geom_in[bn * GIN_ + (t - 64)];
  X1[bn * HID_ + t] = v;
}

// ---------------------------------------------------------------------------
// Gate kernel: one block per (b,e).  h1 = silu(pre1(b,n,:) + pre2(e,:)) in
// LDS; WMMA 64x128 @ 128x128 (layer 2); folds the 128->1 head + sigmoid +
// cosine window, reduces gate and its sum.  8 waves x 4 tiles.
// ---------------------------------------------------------------------------
__global__ void gate_kernel(const float* __restrict__ pre1,   // (B*N,128)
                            const float* __restrict__ pre2,   // (NE,128) incl b0
                            const float* __restrict__ W1,     // (128,128)
                            const float* __restrict__ b1,     // (128)
                            const float* __restrict__ W2,     // (128,1)
                            const float* __restrict__ b2,     // (1)
                            const float* __restrict__ cwv,    // (B*N)
                            float* __restrict__ gate,         // (B*NE,N)
                            float* __restrict__ gsum) {       // (B*NE)
  __shared__ float h1s[N_ * HID_];   // 32 KB
  __shared__ float gpart[N_];
  const int be = blockIdx.x;
  const int b  = be >> 7;            // NE_ = 128
  const int e  = be & 127;
  const int tid = threadIdx.x;       // 256 threads = 8 waves

  for (int idx = tid; idx < N_ * HID_; idx += 256) {
    const int n = idx >> 7, h = idx & 127;
    h1s[idx] = silu_(pre1[(b * N_ + n) * HID_ + h] + pre2[e * HID_ + h]);
  }
  if (tid < N_) gpart[tid] = 0.0f;
  __syncthreads();

  const int wave = tid >> 5;
  const int lane = tid & 31;
  const int m    = lane & 15;
  const int koff = (lane >> 4) << 1;
  for (int tile = wave; tile < 32; tile += 8) {
    const int rt = tile >> 3;        // 4 row tiles (N_=64)
    const int ct = tile & 7;         // 8 col tiles (HID_=128)
    const int col = ct * 16 + m;
    const float* hp = &h1s[(rt * 16 + m) * HID_ + koff];  // 8B-aligned pairs
    const float* wp = W1 + koff * HID_ + col;
    v8f acc = {0.f, 0.f, 0.f, 0.f, 0.f, 0.f, 0.f, 0.f};
    #pragma unroll 8
    for (int k0 = 0; k0 < HID_; k0 += 4) {
      const v2f a = *(const v2f*)hp;
      v2f w;
      w.x = wp[0];
      w.y = wp[HID_];
      acc = __builtin_amdgcn_wmma_f32_16x16x4_f32(false, a, false, w,
                                                  (short)0, acc, false, false);
      hp += 4;
      wp += 4 * HID_;
    }
    const float bb  = b1[col];
    const float w2c = W2[col];
    const int rbase = rt * 16 + ((lane >> 4) << 3);
    #pragma unroll
    for (int j = 0; j < 8; ++j) {
      const float h2 = silu_(acc[j] + bb);
      atomicAdd(&gpart[rbase + j], h2 * w2c);
    }
  }
  __syncthreads();

  if (tid < N_) {
    const float g = sigmoid_(gpart[tid] + b2[0]) * cwv[b * N_ + tid];
    gate[be * N_ + tid] = g;
    gpart[tid] = g;
  }
  __syncthreads();
  if (tid == 0) {
    float s = 0.0f;
    for (int n = 0; n < N_; ++n) s += gpart[n];
    gsum[be] = s;
  }
}

// ---------------------------------------------------------------------------
// PQ kernel: per (b,n) build [P(64) | geom_c(r)*Q(r,64)] from base_w,
// rank_basis and the equivariant features (16x16 matvecs, low-rank algebra).
// ---------------------------------------------------------------------------
__global__ void pq_kernel(const float* __restrict__ h_full,
                          const float* __restrict__ sh1,
                          const float* __restrict__ base_w,   // (B*N,1024)
                          const float* __restrict__ rb,       // (8,1024)
                          const float* __restrict__ geom_c,   // (B*N,8)
                          float* __restrict__ PQ) {           // (B*N,576)
  __shared__ float s1s[MUL_], vds[MUL_], v1s[MUL_ * 3], sh1s[3], gcs[RANK_];
  const int bn  = blockIdx.x;
  const int tid = threadIdx.x;          // 64 threads
  if (tid < MUL_)      s1s[tid] = h_full[bn * DN_ + tid];
  if (tid < MUL_ * 3)  v1s[tid] = h_full[bn * DN_ + MUL_ + tid];
  if (tid < 3)         sh1s[tid] = sh1[bn * 3 + tid];
  if (tid < RANK_)     gcs[tid] = geom_c[bn * RANK_ + tid];
  __syncthreads();
  if (tid < MUL_)
    vds[tid] = v1s[tid * 3] * sh1s[0] + v1s[tid * 3 + 1] * sh1s[1] +
               v1s[tid * 3 + 2] * sh1s[2];
  __syncthreads();

  const float cc = rsqrtf(2.0f * MUL_);
  const float c3 = cc * rsqrtf(3.0f);
  const float* bw = base_w + bn * WNUM_;
  float* out = PQ + bn * PQC_;

  if (tid < MUL_) {                       // out_s channel d = tid
    const int wp = tid;
    float accP = 0.0f;
    for (int u = 0; u < MUL_; ++u)
      accP += cc * bw[u * 16 + wp] * s1s[u] + c3 * bw[768 + u * 16 + wp] * vds[u];
    out[tid] = accP;
    for (int r = 0; r < RANK_; ++r) {
      const float* rbw = rb + r * WNUM_;
      float q = 0.0f;
      for (int u = 0; u < MUL_; ++u)
        q += cc * rbw[u * 16 + wp] * s1s[u] + c3 * rbw[768 + u * 16 + wp] * vds[u];
      out[LAT_ + r * LAT_ + tid] = gcs[r] * q;
    }
  } else {                                // out_v channel d = 16 + 3*wp + i
    const int t  = tid - MUL_;
    const int wp = t / 3;
    const int i  = t - 3 * wp;
    float accP = 0.0f;
    for (int u = 0; u < MUL_; ++u)
      accP += cc * (bw[256 + u * 16 + wp] * s1s[u] * sh1s[i] +
                    bw[512 + u * 16 + wp] * v1s[u * 3 + i]);
    out[tid] = accP;
    for (int r = 0; r < RANK_; ++r) {
      const float* rbw = rb + r * WNUM_;
      float q = 0.0f;
      for (int u = 0; u < MUL_; ++u)
        q += cc * (rbw[256 + u * 16 + wp] * s1s[u] * sh1s[i] +
                   rbw[512 + u * 16 + wp] * v1s[u * 3 + i]);
      out[LAT_ + r * LAT_ + tid] = gcs[r] * q;
    }
  }
}

// ---------------------------------------------------------------------------
// Finalize: agg = (aggP + sum_r en_c*aggQ) / max(gate_sum,1e-8); invariants.
// ---------------------------------------------------------------------------
__global__ void finalize_kernel(const float* __restrict__ aggpre, // (B*NE,576)
                                const float* __restrict__ en_c,   // (NE,8)
                                const float* __restrict__ gsum,   // (B*NE)
                                float* __restrict__ inv_agg) {    // (B*NE,32)
  __shared__ float aggs[LAT_];
  const int row = blockIdx.x;
  const int e   = row & (NE_ - 1);
  const int t   = threadIdx.x;          // 64 threads
  float a = aggpre[row * PQC_ + t];
  for (int r = 0; r < RANK_; ++r)
    a += en_c[e * RANK_ + r] * aggpre[row * PQC_ + LAT_ + r * LAT_ + t];
  a /= fmaxf(gsum[row], 1e-8f);
  aggs[t] = a;
  __syncthreads();
  if (t < 32) {
    float v;
    if (t < MUL_) v = aggs[t];
    else {
      const int mm = t - MUL_;
      const float v0 = aggs[MUL_ + mm * 3 + 0];
      const float v1 = aggs[MUL_ + mm * 3 + 1];
      const float v2 = aggs[MUL_ + mm * 3 + 2];
      v = sqrtf((v0 * v0 + v1 * v1 + v2 * v2) * (1.0f / 3.0f) + 1e-8f);
    }
    inv_agg[row * 32 + t] = v;
  }
}

// ---------------------------------------------------------------------------
extern "C" void kernel_launch(void* const* d_in, const int* in_sizes, int n_in,
                              void* d_out, int out_size, void* d_ws, size_t ws_size,
                              hipStream_t stream) {
  const float* h_full = (const float*)d_in[0];
  const int*   z      = (const int*)d_in[1];
  const float* pos    = (const float*)d_in[2];
  const unsigned char* mask = (const unsigned char*)d_in[3];
  const float* e_feat = (const float*)d_in[4];
  const int*   absp   = (const int*)d_in[5];
  const float* z_emb  = (const float*)d_in[6];
  const float* bwW0 = (const float*)d_in[7],  *bwB0 = (const float*)d_in[8];
  const float* bwW1 = (const float*)d_in[9],  *bwB1 = (const float*)d_in[10];
  const float* bwW2 = (const float*)d_in[11], *bwB2 = (const float*)d_in[12];
  const float* gcW0 = (const float*)d_in[13], *gcB0 = (const float*)d_in[14];
  const float* gcW1 = (const float*)d_in[15], *gcB1 = (const float*)d_in[16];
  const float* gcW2 = (const float*)d_in[17], *gcB2 = (const float*)d_in[18];
  const float* ecW0 = (const float*)d_in[19], *ecB0 = (const float*)d_in[20];
  const float* ecW1 = (const float*)d_in[21], *ecB1 = (const float*)d_in[22];
  const float* ecW2 = (const float*)d_in[23], *ecB2 = (const float*)d_in[24];
  const float* scW0 = (const float*)d_in[25], *scB0 = (const float*)d_in[26];
  const float* scW1 = (const float*)d_in[27], *scB1 = (const float*)d_in[28];
  const float* scW2 = (const float*)d_in[29], *scB2 = (const float*)d_in[30];
  const float* omW0 = (const float*)d_in[31], *omB0 = (const float*)d_in[32];
  const float* omW1 = (const float*)d_in[33], *omB1 = (const float*)d_in[34];
  const float* omW2 = (const float*)d_in[35], *omB2 = (const float*)d_in[36];
  const float* rbasis = (const float*)d_in[37];
  float* out = (float*)d_out;

  // workspace layout (floats)
  float* ws = (float*)d_ws;
  const int BN = B_ * N_;                 // 512
  const int BE = B_ * NE_;                // 1024
  float* geom  = ws;                      long long o = (long long)BN * GIN_;
  float* inv   = ws + o;                  o += (long long)BN * 32;
  float* sh1   = ws + o;                  o += (long long)BN * 3;
  float* cwv   = ws + o;                  o += BN;
  float* X1    = ws + o;                  o += (long long)BN * HID_;
  float* t1    = ws + o;                  o += (long long)BE * HID_;  // generous scratch
  float* t2    = ws + o;                  o += (long long)BE * HID_;
  float* basew = ws + o;                  o += (long long)BN * WNUM_;
  float* geomc = ws + o;                  o += (long long)BN * RANK_;
  float* enc   = ws + o;                  o += (long long)NE_ * RANK_;
  float* pre1  = ws + o;                  o += (long long)BN * HID_;
  float* pre2  = ws + o;                  o += (long long)NE_ * HID_;
  float* PQ    = ws + o;                  o += (long long)BN * PQC_;
  float* gate  = ws + o;                  o += (long long)BE * N_;
  float* gsum  = ws + o;                  o += BE;
  float* aggp  = ws + o;                  o += (long long)BE * PQC_;
  float* iagg  = ws + o;                  o += (long long)BE * 32;
  (void)ws_size; (void)in_sizes; (void)n_in; (void)out_size;

  auto gemm = [&](const float* A, const float* W, const float* bias, float* C,
                  int R, int K, int Cn, int act, int batches = 1,
                  long long sA = 0, long long sB = 0, long long sC = 0) {
    dim3 grid((R + 15) / 16, (Cn + 15) / 16, batches);
    gemm_bias_act<<<grid, 32, 0, stream>>>(A, W, bias, C, R, K, Cn, sA, sB, sC, act);
  };

  // geometry + per-(b,n) features
  geom_kernel<<<BN, 32, 0, stream>>>(h_full, z, pos, mask, absp, z_emb,
                                     geom, inv, sh1, cwv);
  x1_kernel<<<BN, HID_, 0, stream>>>(inv, geom, absp, X1);

  // base_w MLP: 64 -> 128 -> 128 -> 1024
  gemm(geom, bwW0, bwB0, t1, BN, GIN_, HID_, 1);
  gemm(t1,   bwW1, bwB1, t2, BN, HID_, HID_, 1);
  gemm(t2,   bwW2, bwB2, basew, BN, HID_, WNUM_, 0);
  // geom_c MLP: 64 -> 128 -> 128 -> 8
  gemm(geom, gcW0, gcB0, t1, BN, GIN_, HID_, 1);
  gemm(t1,   gcW1, gcB1, t2, BN, HID_, HID_, 1);
  gemm(t2,   gcW2, gcB2, geomc, BN, HID_, RANK_, 0);
  // energy_c MLP: 16 -> 128 -> 128 -> 8
  gemm(e_feat, ecW0, ecB0, t1, NE_, DE_, HID_, 1);
  gemm(t1,     ecW1, ecB1, t2, NE_, HID_, HID_, 1);
  gemm(t2,     ecW2, ecB2, enc, NE_, HID_, RANK_, 0);
  // score layer-1 split: pre1 = X1 @ W0[:128,:]; pre2 = e_feat @ W0[128:,:] + b0
  gemm(X1, scW0, nullptr, pre1, BN, HID_, HID_, 0);
  gemm(e_feat, scW0 + 128 * HID_, scB0, pre2, NE_, DE_, HID_, 0);
  // fused gate (layer-2 WMMA + head + window + normalizer sum)
  gate_kernel<<<BE, 256, 0, stream>>>(pre1, pre2, scW1, scB1, scW2, scB2,
                                      cwv, gate, gsum);
  // low-rank value factors
  pq_kernel<<<BN, 64, 0, stream>>>(h_full, sh1, basew, rbasis, geomc, PQ);
  // batched aggregation GEMM: per b, gate(128x64) @ PQ(64x576)
  gemm(gate, PQ, nullptr, aggp, NE_, N_, PQC_, 0, B_,
       (long long)NE_ * N_, (long long)N_ * PQC_, (long long)NE_ * PQC_);
  // rank contraction + normalize + invariants
  finalize_kernel<<<BE, 64, 0, stream>>>(aggp, enc, gsum, iagg);
  // out MLP: 32 -> 128 -> 128 -> 64 (writes d_out)
  gemm(iagg, omW0, omB0, t1, BE, 2 * MUL_, HID_, 1);
  gemm(t1,   omW1, omB1, t2, BE, HID_, HID_, 1);
  gemm(t2,   omW2, omB2, out, BE, HID_, LAT_, 0);
}